// EatRxnLayer_63239098466494
// MI455X (gfx1250) — compile-verified
//
#include <hip/hip_runtime.h>

// EatRxnLayer: output = [muTE (N floats), M (N*N floats)], fp32, N=8192.
// M is zero except rows/cols 0,1 -> cost is writing 256MB of zeros (~11us at
// 23.3 TB/s HBM) + ~4N nonzero entries. Fill uses the gfx1250 async
// LDS->global store path (ASYNCcnt) when available, else NT b128 stores.

typedef float v4f __attribute__((ext_vector_type(4)));
typedef float v2f __attribute__((ext_vector_type(2)));
typedef int   v4i __attribute__((vector_size(4 * sizeof(int))));

#if __has_builtin(__builtin_amdgcn_global_store_async_from_lds_b128)
#define FILL_ASYNC 1
typedef __attribute__((address_space(1))) v4i* gas_v4i_ptr;  // global
typedef __attribute__((address_space(3))) v4i* las_v4i_ptr;  // LDS
#else
#define FILL_ASYNC 0
#endif

// Zero-fill: each thread handles one 64B chunk (4 x b128) per grid-stride step.
__global__ void fill_zero(float* __restrict__ out, long long n64B, long long totalFloats) {
    long long i      = (long long)blockIdx.x * blockDim.x + threadIdx.x;
    long long stride = (long long)gridDim.x * blockDim.x;

#if FILL_ASYNC
    // 64B of zeros in LDS; async b128 stores read LDS at (base + INST_OFFSET),
    // so the buffer must cover offsets 0..48.
    __shared__ __align__(64) float zbuf[16];
    if (threadIdx.x < 16) zbuf[threadIdx.x] = 0.0f;
    __syncthreads();
    las_v4i_ptr lds = (las_v4i_ptr)zbuf;
    for (; i < n64B; i += stride) {
        gas_v4i_ptr g = (gas_v4i_ptr)(out + i * 16);
        __builtin_amdgcn_global_store_async_from_lds_b128(g, lds,  0, 0);
        __builtin_amdgcn_global_store_async_from_lds_b128(g, lds, 16, 0);
        __builtin_amdgcn_global_store_async_from_lds_b128(g, lds, 32, 0);
        __builtin_amdgcn_global_store_async_from_lds_b128(g, lds, 48, 0);
    }
#if __has_builtin(__builtin_amdgcn_s_wait_asynccnt)
    __builtin_amdgcn_s_wait_asynccnt(0);
#endif
#else
    v4f z = {0.0f, 0.0f, 0.0f, 0.0f};
    for (; i < n64B; i += stride) {
        v4f* p = (v4f*)(out + i * 16);
        __builtin_nontemporal_store(z, p + 0);
        __builtin_nontemporal_store(z, p + 1);
        __builtin_nontemporal_store(z, p + 2);
        __builtin_nontemporal_store(z, p + 3);
    }
#endif

    // scalar tail if totalFloats not a multiple of 16 (not hit for N=8192)
    if (blockIdx.x == 0 && threadIdx.x == 0) {
        for (long long t = n64B * 16; t < totalFloats; ++t) {
            __builtin_nontemporal_store(0.0f, out + t);
        }
    }
}

__global__ void fixup(const float* __restrict__ mu,
                      const float* __restrict__ var,
                      float* __restrict__ muTE,
                      float* __restrict__ M,
                      int n) {
    int j = blockIdx.x * blockDim.x + threadIdx.x;
    if (j >= n) return;

    const long long nn = (long long)n;

    // broadcast scalars (h = 0, p = 1)
    const float mh  = mu[0];
    const float mp  = mu[1];
    const float vhp = var[1];        // var[h,p]
    const float vph = var[nn];       // var[p,h]
    const float hp  = vhp;           // reference: nvar[h,p]

    // v[j] = mu[j]*(-2*mp*mh + vph) + mp*var[j,h] + mh*var[j,p]
    const float c0 = -2.0f * mp * mh + vph;
    long long row = (long long)j * nn;
    v2f vjcols = *(const v2f*)(var + row);   // {var[j,0], var[j,1]} (8B aligned: n even)
    float v = mu[j] * c0 + mp * vjcols.x + mh * vjcols.y;

    if (j >= 2) {
        // row h and row p
        M[j]      =  v;      // M[0, j]
        M[nn + j] = -v;      // M[1, j]
        // column h and column p: one 8-byte store {v, -v} at M[j, 0]
        v2f pairv = { v, -v };
        *(v2f*)(M + row) = pairv;
    } else {
        // special scalars (computed redundantly by threads 0 and 1; cheap)
        const float vhh = var[0];
        const float vpp = var[nn + 1];
        const float n3_pph = -2.0f * mp * mp * mh + 2.0f * mp * vph + mh * vpp; // _n3(p,p,h)
        const float n3_hhp = -2.0f * mh * mh * mp + 2.0f * mh * vhp + mp * vhh; // _n3(h,h,p)
        const float n3_hpp = -2.0f * mh * mp * mp + mh * vpp + 2.0f * mp * vhp; // _n3(h,p,p)
        const float chp = -n3_hhp + n3_hpp - hp;
        if (j == 0) {
            muTE[0] = hp;
            M[0]    = 2.0f * n3_hhp + hp;    // M[h,h]
            M[nn]   = chp;                   // M[p,h]
        } else { // j == 1
            muTE[1]   = -hp;
            M[1]      = chp;                 // M[h,p]
            M[nn + 1] = -2.0f * n3_pph + hp; // M[p,p]
        }
    }
}

extern "C" void kernel_launch(void* const* d_in, const int* in_sizes, int n_in,
                              void* d_out, int out_size, void* d_ws, size_t ws_size,
                              hipStream_t stream) {
    const float* mu  = (const float*)d_in[0];
    const float* var = (const float*)d_in[1];
    float* out = (float*)d_out;

    const long long n     = (long long)in_sizes[0];     // 8192
    const long long total = n + n * n;                  // muTE + M
    const long long n64B  = total / 16;                 // 64-byte chunks (exact for N=8192)

    float* muTE = out;
    float* M    = out + n;

    const int threads = 256;                            // 8 waves (wave32)
    // ~4.2M 64B chunks; 4096 blocks * 256 threads -> 4 chunks (16 b128 stores)/thread
    fill_zero<<<4096, threads, 0, stream>>>(out, n64B, total);

    const int fixBlocks = (int)((n + threads - 1) / threads);
    fixup<<<fixBlocks, threads, 0, stream>>>(mu, var, muTE, M, (int)n);
}